// DQNNetwork_28887950033239
// MI455X (gfx1250) — compile-verified
//
#include <hip/hip_runtime.h>

typedef __attribute__((ext_vector_type(2))) float v2f;
typedef __attribute__((ext_vector_type(8))) float v8f;

#define Bg   512
#define OBSD 256
#define NPG  128
#define EPG  1024
#define SW   200
#define NAD  256
#define FIN  5
#define HD   128
#define NTOT (Bg*NPG)
#define ETOT (Bg*EPG)
#define NEGV (-100000000.0f)

__device__ __forceinline__ v8f wmma_f32(v2f a, v2f b, v8f c) {
  // D = A(16x4,f32) * B(4x16,f32) + C(16x16,f32)
  return __builtin_amdgcn_wmma_f32_16x16x4_f32(false, a, false, b, (short)0, c,
                                               false, false);
}

// ---------------- Layer 1: per-graph degree + mean-agg + tiny GEMM ----------
__global__ void k1_layer1(const float* __restrict__ x,
                          const int* __restrict__ srcI,
                          const int* __restrict__ dstI,
                          const float* __restrict__ Ws1,
                          const float* __restrict__ bs1,
                          const float* __restrict__ Wn1,
                          float* __restrict__ h1,
                          float* __restrict__ deg_inv) {
  __shared__ float deg[NPG];
  __shared__ float agg[NPG * FIN];
  __shared__ float xs[NPG * FIN];
  __shared__ float w1[FIN * HD];
  __shared__ float w2[FIN * HD];
  __shared__ float b1[HD];
  const int g = blockIdx.x;
  const int t = threadIdx.x;
  for (int i = t; i < NPG; i += 256) deg[i] = 0.f;
  for (int i = t; i < NPG * FIN; i += 256) {
    agg[i] = 0.f;
    xs[i] = x[(size_t)g * NPG * FIN + i];
  }
  for (int i = t; i < FIN * HD; i += 256) { w1[i] = Ws1[i]; w2[i] = Wn1[i]; }
  for (int i = t; i < HD; i += 256) b1[i] = bs1[i];
  __syncthreads();
  for (int e = t; e < EPG; e += 256) {
    int s = srcI[(size_t)g * EPG + e] - g * NPG;
    int d = dstI[(size_t)g * EPG + e] - g * NPG;
    atomicAdd(&deg[d], 1.0f);
#pragma unroll
    for (int c = 0; c < FIN; ++c) atomicAdd(&agg[d * FIN + c], xs[s * FIN + c]);
  }
  __syncthreads();
  if (t < NPG) {
    float dv = 1.0f / fmaxf(deg[t], 1.0f);
    deg[t] = dv;
    deg_inv[(size_t)g * NPG + t] = dv;
  }
  __syncthreads();
  for (int i = t; i < NPG * HD; i += 256) {
    int n = i >> 7, j = i & (HD - 1);
    float dv = deg[n];
    float sum = b1[j];
#pragma unroll
    for (int c = 0; c < FIN; ++c)
      sum += xs[n * FIN + c] * w1[c * HD + j] +
             agg[n * FIN + c] * dv * w2[c * HD + j];
    h1[((size_t)g * NPG + n) * HD + j] = fmaxf(sum, 0.f);
  }
}

// ---------------- Layer 2: per-graph LDS scatter-agg + fused WMMA GEMM ------
__global__ void k2_layer2(const float* __restrict__ h1,
                          const int* __restrict__ srcI,
                          const int* __restrict__ dstI,
                          const float* __restrict__ Wcat,  // [2H][H] = [Ws2;Wn2]
                          const float* __restrict__ bs2,
                          const float* __restrict__ deg_inv,
                          float* __restrict__ h2) {
  extern __shared__ float aggH[];  // NPG*HD floats = 64 KB
  const int g = blockIdx.x;
  const int t = threadIdx.x;
  for (int i = t; i < NPG * HD; i += 256) aggH[i] = 0.f;
  __syncthreads();
  // edge scatter: E*32 chunk-units of 4 channels each -> ds_add_f32
  for (int u = t; u < EPG * (HD / 4); u += 256) {
    int e = u >> 5;
    int cq = (u & 31) << 2;
    int sg = srcI[(size_t)g * EPG + e];
    int dl = dstI[(size_t)g * EPG + e] - g * NPG;
    const float4 hv = *reinterpret_cast<const float4*>(&h1[(size_t)sg * HD + cq]);
    atomicAdd(&aggH[dl * HD + cq + 0], hv.x);
    atomicAdd(&aggH[dl * HD + cq + 1], hv.y);
    atomicAdd(&aggH[dl * HD + cq + 2], hv.z);
    atomicAdd(&aggH[dl * HD + cq + 3], hv.w);
  }
  __syncthreads();
  // fused GEMM: [h1_g | agg2] (128x256) @ Wcat (256x128), wave-per-16-rows
  const int lane = t & 31;
  const int wave = t >> 5;
  const int lm = lane & 15;
  const int kh = lane >> 4;
  const int mrow = wave * 16 + lm;
  const float dinv = deg_inv[(size_t)g * NPG + mrow];
  const float* h1row = &h1[((size_t)g * NPG + mrow) * HD];
  __builtin_prefetch(h1row, 0, 3);
  v8f acc[8] = {};
  // K half 1: A from h1 (global, contiguous b64 per lane)
  for (int ks = 0; ks < HD; ks += 4) {
    const int k = ks + 2 * kh;
    v2f a;
    a.x = h1row[k];
    a.y = h1row[k + 1];
    const float* w0 = &Wcat[(size_t)k * HD + lm];
    const float* w1 = &Wcat[(size_t)(k + 1) * HD + lm];
#pragma unroll
    for (int nt = 0; nt < 8; ++nt) {
      v2f b;
      b.x = w0[nt * 16];
      b.y = w1[nt * 16];
      acc[nt] = wmma_f32(a, b, acc[nt]);
    }
  }
  // K half 2: A from LDS aggregation, scaled by 1/deg (mean aggregation)
  for (int ks = 0; ks < HD; ks += 4) {
    const int k = ks + 2 * kh;
    v2f a;
    a.x = aggH[mrow * HD + k] * dinv;
    a.y = aggH[mrow * HD + k + 1] * dinv;
    const float* w0 = &Wcat[(size_t)(HD + k) * HD + lm];
    const float* w1 = &Wcat[(size_t)(HD + k + 1) * HD + lm];
#pragma unroll
    for (int nt = 0; nt < 8; ++nt) {
      v2f b;
      b.x = w0[nt * 16];
      b.y = w1[nt * 16];
      acc[nt] = wmma_f32(a, b, acc[nt]);
    }
  }
#pragma unroll
  for (int nt = 0; nt < 8; ++nt) {
#pragma unroll
    for (int r = 0; r < 8; ++r) {
      int row = wave * 16 + r + 8 * kh;
      int col = nt * 16 + lm;
      float v = acc[nt][r] + bs2[col];
      h2[((size_t)g * NPG + row) * HD + col] = fmaxf(v, 0.f);
    }
  }
}

// ---------------- Graph pooling (mean|max) into fusion buffer ---------------
__global__ void k3_pool(const float* __restrict__ h2, float* __restrict__ fused) {
  const int g = blockIdx.x;
  const int c = threadIdx.x;  // 128 threads
  float s = 0.f, mx = -3.4e38f;
  for (int n = 0; n < NPG; ++n) {
    float v = h2[((size_t)g * NPG + n) * HD + c];
    s += v;
    mx = fmaxf(mx, v);
  }
  fused[(size_t)g * 384 + HD + c] = s * (1.0f / NPG);
  fused[(size_t)g * 384 + 2 * HD + c] = mx;
}

// ------ Generic WMMA GEMM: wave computes 16 rows x 64 cols (4 N-tiles) ------
__global__ void gemm_wmma4(const float* __restrict__ A, int lda,
                           const float* __restrict__ W, int ldw,
                           const float* __restrict__ bias,
                           const float* __restrict__ res, int ldres,
                           float* __restrict__ C, int ldc, int K, int doRelu) {
  const int nb4 = blockIdx.x;  // group of 4 N-tiles (64 columns)
  const int mt = blockIdx.y;
  const int lane = threadIdx.x;
  const int lm = lane & 15, kh = lane >> 4;
  const int m = mt * 16 + lm;
  const int ncol = nb4 * 64 + lm;
  const float* arow = A + (size_t)m * lda;
  __builtin_prefetch(arow, 0, 3);
  v8f acc[4] = {};
  for (int ks = 0; ks < K; ks += 4) {
    const int k = ks + 2 * kh;
    v2f a;
    a.x = arow[k];
    a.y = arow[k + 1];
    const float* w0 = W + (size_t)k * ldw + ncol;
    const float* w1 = W + (size_t)(k + 1) * ldw + ncol;
#pragma unroll
    for (int nt = 0; nt < 4; ++nt) {
      v2f b;
      b.x = w0[nt * 16];
      b.y = w1[nt * 16];
      acc[nt] = wmma_f32(a, b, acc[nt]);
    }
  }
#pragma unroll
  for (int nt = 0; nt < 4; ++nt) {
#pragma unroll
    for (int r = 0; r < 8; ++r) {
      int row = mt * 16 + r + 8 * kh;
      int col = nb4 * 64 + nt * 16 + lm;
      float v = acc[nt][r];
      if (bias) v += bias[col];
      if (res) v += res[(size_t)row * ldres + col];
      if (doRelu) v = fmaxf(v, 0.f);
      C[(size_t)row * ldc + col] = v;
    }
  }
}

// ---------------- Row LayerNorm + ReLU (in place, strided rows) -------------
__global__ void ln_relu(float* __restrict__ base, int stride, int L,
                        const float* __restrict__ gam,
                        const float* __restrict__ bet) {
  __shared__ float red[256];
  const int row = blockIdx.x;
  const int t = threadIdx.x;
  float* p = base + (size_t)row * stride;
  float xv = (t < L) ? p[t] : 0.f;
  red[t] = xv;
  __syncthreads();
  for (int s = 128; s > 0; s >>= 1) {
    if (t < s) red[t] += red[t + s];
    __syncthreads();
  }
  float mean = red[0] / (float)L;
  __syncthreads();
  float d = (t < L) ? (xv - mean) : 0.f;
  red[t] = d * d;
  __syncthreads();
  for (int s = 128; s > 0; s >>= 1) {
    if (t < s) red[t] += red[t + s];
    __syncthreads();
  }
  float rstd = rsqrtf(red[0] / (float)L + 1e-5f);
  if (t < L) p[t] = fmaxf(d * rstd * gam[t] + bet[t], 0.f);
}

// ---------------- Edge-Q head: wave-per-(b,s) gather + dot ------------------
__global__ void k_edge(const float* __restrict__ na, const float* __restrict__ nb,
                       const float* __restrict__ cc,
                       const int* __restrict__ swp,
                       const float* __restrict__ eq2w,
                       const float* __restrict__ eq2b,
                       float* __restrict__ out) {
  const int wid = blockIdx.x * 8 + (threadIdx.x >> 5);
  const int lane = threadIdx.x & 31;
  const int b = wid / SW;
  const int s = wid % SW;
  const int ns = swp[((size_t)b * SW + s) * 2 + 0];
  const int nd = swp[((size_t)b * SW + s) * 2 + 1];
  const float* pa = &na[((size_t)b * NPG + ns) * HD];
  const float* pb = &nb[((size_t)b * NPG + nd) * HD];
  const float* pc = &cc[(size_t)b * HD];
  float part = 0.f;
#pragma unroll
  for (int c0 = 0; c0 < HD; c0 += 32) {
    int c = c0 + lane;
    float v = fmaxf(pa[c] + pb[c] + pc[c], 0.f);
    part += v * eq2w[c];
  }
  for (int off = 16; off > 0; off >>= 1) part += __shfl_down(part, off, 32);
  if (lane == 0) out[(size_t)b * NAD + s] = part + eq2b[0];
}

// ---------------- Pass-Q + NEG padding --------------------------------------
__global__ void k_pq_pad(const float* __restrict__ ctx,
                         const float* __restrict__ pq1w,
                         const float* __restrict__ pq1b,
                         const float* __restrict__ pq2w,
                         const float* __restrict__ pq2b,
                         float* __restrict__ out) {
  __shared__ float red[64];
  const int b = blockIdx.x;
  const int j = threadIdx.x;  // 64 threads
  const float* pc = &ctx[(size_t)b * 256];
  float u = pq1b[j];
  for (int k = 0; k < 256; ++k) u += pc[k] * pq1w[k * 64 + j];
  u = fmaxf(u, 0.f);
  red[j] = u * pq2w[j];
  __syncthreads();
  for (int s2 = 32; s2 > 0; s2 >>= 1) {
    if (j < s2) red[j] += red[j + s2];
    __syncthreads();
  }
  if (j == 0) out[(size_t)b * NAD + SW] = red[0] + pq2b[0];
  if (j < NAD - SW - 1) out[(size_t)b * NAD + SW + 1 + j] = NEGV;
}

extern "C" void kernel_launch(void* const* d_in, const int* in_sizes, int n_in,
                              void* d_out, int out_size, void* d_ws,
                              size_t ws_size, hipStream_t stream) {
  (void)in_sizes; (void)n_in; (void)out_size; (void)ws_size;
  const float* obs  = (const float*)d_in[0];
  const float* x    = (const float*)d_in[1];
  const int*   ei   = (const int*)d_in[2];
  const int*   swp  = (const int*)d_in[3];
  const float* Ws1  = (const float*)d_in[4];
  const float* bs1  = (const float*)d_in[5];
  const float* Wn1  = (const float*)d_in[6];
  const float* Ws2  = (const float*)d_in[7];
  const float* bs2  = (const float*)d_in[8];
  const float* Wn2  = (const float*)d_in[9];
  const float* o1w  = (const float*)d_in[10];
  const float* o1b  = (const float*)d_in[11];
  const float* o1g  = (const float*)d_in[12];
  const float* o1bb = (const float*)d_in[13];
  const float* o2w  = (const float*)d_in[14];
  const float* o2b  = (const float*)d_in[15];
  const float* o2g  = (const float*)d_in[16];
  const float* o2bb = (const float*)d_in[17];
  const float* gfw  = (const float*)d_in[18];
  const float* gfb  = (const float*)d_in[19];
  const float* gfg  = (const float*)d_in[20];
  const float* gfbb = (const float*)d_in[21];
  const float* r1w  = (const float*)d_in[22];
  const float* r1b  = (const float*)d_in[23];
  const float* r2w  = (const float*)d_in[24];
  const float* r2b  = (const float*)d_in[25];
  const float* eq1w = (const float*)d_in[26];
  const float* eq1b = (const float*)d_in[27];
  const float* eq2w = (const float*)d_in[28];
  const float* eq2b = (const float*)d_in[29];
  const float* pq1w = (const float*)d_in[30];
  const float* pq1b = (const float*)d_in[31];
  const float* pq2w = (const float*)d_in[32];
  const float* pq2b = (const float*)d_in[33];
  float* out = (float*)d_out;

  float* w = (float*)d_ws;
  float* deg_inv = w; w += NTOT;
  float* h1 = w;      w += (size_t)NTOT * HD;
  float* h2 = w;      w += (size_t)NTOT * HD;
  float* na = w;      w += (size_t)NTOT * HD;
  float* nb = w;      w += (size_t)NTOT * HD;
  float* Wcat = w;    w += 2 * HD * HD;
  float* fused = w;   w += Bg * 384;
  float* t1 = w;      w += Bg * HD;
  float* ctx = w;     w += Bg * 2 * HD;
  float* t2 = w;      w += Bg * 2 * HD;
  float* cc = w;      w += Bg * HD;

  const int* srcI = ei;
  const int* dstI = ei + ETOT;

  // Wcat = [Ws2 ; Wn2]  (rows = K dim)
  hipMemcpyAsync(Wcat, Ws2, HD * HD * sizeof(float), hipMemcpyDeviceToDevice, stream);
  hipMemcpyAsync(Wcat + HD * HD, Wn2, HD * HD * sizeof(float), hipMemcpyDeviceToDevice, stream);

  k1_layer1<<<Bg, 256, 0, stream>>>(x, srcI, dstI, Ws1, bs1, Wn1, h1, deg_inv);
  k2_layer2<<<Bg, 256, NPG * HD * sizeof(float), stream>>>(h1, srcI, dstI, Wcat,
                                                           bs2, deg_inv, h2);
  k3_pool<<<Bg, 128, 0, stream>>>(h2, fused);

  dim3 blk(32);
  // obs encoder
  gemm_wmma4<<<dim3(HD / 64, Bg / 16), blk, 0, stream>>>(obs, OBSD, o1w, HD, o1b,
                                                         nullptr, 0, t1, HD, OBSD, 0);
  ln_relu<<<Bg, 256, 0, stream>>>(t1, HD, HD, o1g, o1bb);
  gemm_wmma4<<<dim3(HD / 64, Bg / 16), blk, 0, stream>>>(t1, HD, o2w, HD, o2b,
                                                         nullptr, 0, fused, 384, HD, 0);
  ln_relu<<<Bg, 256, 0, stream>>>(fused, 384, HD, o2g, o2bb);
  // fusion + residual MLP
  gemm_wmma4<<<dim3(2 * HD / 64, Bg / 16), blk, 0, stream>>>(fused, 384, gfw, 2 * HD,
                                                             gfb, nullptr, 0, ctx,
                                                             2 * HD, 384, 0);
  ln_relu<<<Bg, 256, 0, stream>>>(ctx, 2 * HD, 2 * HD, gfg, gfbb);
  gemm_wmma4<<<dim3(2 * HD / 64, Bg / 16), blk, 0, stream>>>(ctx, 2 * HD, r1w, 2 * HD,
                                                             r1b, nullptr, 0, t2,
                                                             2 * HD, 2 * HD, 1);
  gemm_wmma4<<<dim3(2 * HD / 64, Bg / 16), blk, 0, stream>>>(t2, 2 * HD, r2w, 2 * HD,
                                                             r2b, ctx, 2 * HD, ctx,
                                                             2 * HD, 2 * HD, 0);
  // factorized edge-Q hidden layer
  gemm_wmma4<<<dim3(HD / 64, NTOT / 16), blk, 0, stream>>>(h2, HD, eq1w, HD, nullptr,
                                                           nullptr, 0, na, HD, HD, 0);
  gemm_wmma4<<<dim3(HD / 64, NTOT / 16), blk, 0, stream>>>(h2, HD, eq1w + HD * HD, HD,
                                                           nullptr, nullptr, 0, nb,
                                                           HD, HD, 0);
  gemm_wmma4<<<dim3(HD / 64, Bg / 16), blk, 0, stream>>>(ctx, 2 * HD,
                                                         eq1w + 2 * HD * HD, HD, eq1b,
                                                         nullptr, 0, cc, HD, 2 * HD, 0);

  k_edge<<<(Bg * SW) / 8, 256, 0, stream>>>(na, nb, cc, swp, eq2w, eq2b, out);
  k_pq_pad<<<Bg, 64, 0, stream>>>(ctx, pq1w, pq1b, pq2w, pq2b, out);
}